// Net_86474871538649
// MI455X (gfx1250) — compile-verified
//
#include <hip/hip_runtime.h>
#include <hip/hip_bf16.h>
#include <math.h>

typedef float v2f __attribute__((ext_vector_type(2)));
typedef float v8f __attribute__((ext_vector_type(8)));

#define LEAKY(v) ((v) > 0.0f ? (v) : 0.01f * (v))

__device__ __forceinline__ void atomAddF(float* p, float v) {
  unsafeAtomicAdd(p, v);   // lowers to global_atomic_add_f32 on gfx1250
}

__device__ __forceinline__ void atomMaxF(float* p, float v) {
  // signed/unsigned monotonic-encoding trick; dst pre-filled with -INF
  if (v >= 0.0f) atomicMax((int*)p, __float_as_int(v));
  else           atomicMin((unsigned int*)p, __float_as_uint(v));
}

// ---------------------------------------------------------------------------
// fp32 GEMM with V_WMMA_F32_16X16X4_F32.
// out[M,Nc] = epilogue( A[M,K] @ W[K,Nc] + bias )
// flags: bit0 -> apply (v*s + t), bit1 -> leaky_relu(0.01)
// One wave computes a 16(M) x 64(N) strip: 4 accumulators share each A frag,
// K stepped by 4. Requires Nc % 64 == 0 (true here: 128/256/64).
// Partial M tiles: clamp load rows (EXEC stays all-ones), guard stores.
// ---------------------------------------------------------------------------
#define WMMA4(cacc, bfrag) \
  cacc = __builtin_amdgcn_wmma_f32_16x16x4_f32(false, a, false, (bfrag), \
                                               (short)0, cacc, false, false)

__global__ void gemm_wmma_kernel(const float* __restrict__ A,
                                 const float* __restrict__ W,
                                 const float* __restrict__ bias,
                                 const float* __restrict__ s,
                                 const float* __restrict__ t,
                                 float* __restrict__ out,
                                 int M, int K, int Nc, int flags) {
  const int lane  = threadIdx.x & 31;
  const int wave  = threadIdx.x >> 5;
  const int nBase = blockIdx.x * 64;                         // 4 n-tiles of 16
  const int mBase = (blockIdx.y * (blockDim.x >> 5) + wave) * 16;
  if (mBase >= M) return;                  // wave-uniform: EXEC full below

  const int half = lane >> 4;              // 0: K pair {0,1}; 1: K pair {2,3}
  const int l15  = lane & 15;

  int mRow = mBase + l15;
  if (mRow >= M) mRow = M - 1;             // clamp loads for last tile
  const float* Arow = A + (size_t)mRow * K + half * 2;       // 8B aligned
  const float* Wcol = W + (size_t)(half * 2) * Nc + nBase + l15;

  v8f c0 = {}, c1 = {}, c2 = {}, c3 = {};
  for (int k0 = 0; k0 < K; k0 += 4) {
    const v2f a = *(const v2f*)Arow;       // one b64 load feeds 4 WMMAs
    Arow += 4;
    v2f b0, b1, b2, b3;
    b0.x = Wcol[0];       b0.y = Wcol[Nc];
    b1.x = Wcol[16];      b1.y = Wcol[Nc + 16];
    b2.x = Wcol[32];      b2.y = Wcol[Nc + 32];
    b3.x = Wcol[48];      b3.y = Wcol[Nc + 48];
    Wcol += (size_t)4 * Nc;
    WMMA4(c0, b0);
    WMMA4(c1, b1);
    WMMA4(c2, b2);
    WMMA4(c3, b3);
  }

#pragma unroll
  for (int jt = 0; jt < 4; ++jt) {
    const v8f c = (jt == 0) ? c0 : (jt == 1) ? c1 : (jt == 2) ? c2 : c3;
    const int col = nBase + jt * 16 + l15;
    const float bcol = bias[col];
    const float sc   = (flags & 1) ? s[col] : 1.0f;
    const float tc   = (flags & 1) ? t[col] : 0.0f;
#pragma unroll
    for (int j = 0; j < 8; ++j) {
      const int row = mBase + j + half * 8;
      if (row < M) {
        float v = c[j] + bcol;
        v = v * sc + tc;
        if (flags & 2) v = LEAKY(v);
        out[(size_t)row * Nc + col] = v;
      }
    }
  }
}

// ---------------------------------------------------------------------------
// Small helpers
// ---------------------------------------------------------------------------
__global__ void fill_kernel(float* p, float v, size_t n) {
  size_t i = (size_t)blockIdx.x * blockDim.x + threadIdx.x;
  if (i < n) p[i] = v;
}

// per-edge weight = edge_attr @ We + be for both layers at once
__global__ void edge_w_kernel(const float* __restrict__ ea,
                              const float* __restrict__ We0, const float* __restrict__ be0,
                              const float* __restrict__ We1, const float* __restrict__ be1,
                              float* __restrict__ w0, float* __restrict__ w1, int E) {
  int e = blockIdx.x * blockDim.x + threadIdx.x;
  if (e >= E) return;
  const float* a = ea + (size_t)e * 12;
  float s0 = be0[0], s1 = be1[0];
#pragma unroll
  for (int i = 0; i < 12; ++i) {
    float v = a[i];
    s0 += v * We0[i];
    s1 += v * We1[i];
  }
  w0[e] = s0;
  w1[e] = s1;
}

__global__ void wself_kernel(const float* We0, const float* be0,
                             const float* We1, const float* be1, float* out) {
  if (threadIdx.x == 0 && blockIdx.x == 0) {
    float s = 0.f;
    for (int i = 0; i < 12; ++i) s += We0[i];
    out[0] = s + be0[0];
    s = 0.f;
    for (int i = 0; i < 12; ++i) s += We1[i];
    out[1] = s + be1[0];
  }
}

// degree accumulation on row (optionally cluster-remapped with self-edge mask)
__global__ void edge_deg_kernel(const int* __restrict__ row, const int* __restrict__ col,
                                const int* __restrict__ cluster, float* __restrict__ deg, int E) {
  int e = blockIdx.x * blockDim.x + threadIdx.x;
  if (e >= E) return;
  int r = row[e];
  if (cluster) {
    int c = cluster[col[e]];
    r = cluster[r];
    if (r == c) return;                   // pooled self-edges are masked out
  }
  atomAddF(&deg[r], 1.0f);
}

__global__ void dis_kernel(float* deg, int n) {
  int i = blockIdx.x * blockDim.x + threadIdx.x;
  if (i < n) deg[i] = rsqrtf(deg[i] + 1.0f);   // +1 self-loop
}

// one wave per edge: lane handles 4 of the 128 features (float4)
__global__ void edge_agg_kernel(const float* __restrict__ h,
                                const int* __restrict__ row, const int* __restrict__ col,
                                const float* __restrict__ w, const float* __restrict__ dis,
                                const int* __restrict__ cluster,
                                float* __restrict__ agg, int E) {
  int gid = blockIdx.x * blockDim.x + threadIdx.x;
  int e = gid >> 5;
  if (e >= E) return;
  int lane = gid & 31;
  int r = row[e], c = col[e];
  if (cluster) {
    r = cluster[r];
    c = cluster[c];
    if (r == c) return;
  }
  float coef = dis[r] * dis[c] * w[e];
  float4 v = ((const float4*)(h + (size_t)r * 128))[lane];
  float* dst = agg + (size_t)c * 128 + lane * 4;
  atomAddF(dst + 0, coef * v.x);
  atomAddF(dst + 1, coef * v.y);
  atomAddF(dst + 2, coef * v.z);
  atomAddF(dst + 3, coef * v.w);
}

// agg = leaky( agg + dis^2 * w_self * h  (+ origin) )
__global__ void combine_kernel(float* __restrict__ agg, const float* __restrict__ hm,
                               const float* __restrict__ origin,
                               const float* __restrict__ dis,
                               const float* __restrict__ wselfp, int wsidx,
                               int Nn, int D) {
  size_t i = (size_t)blockIdx.x * blockDim.x + threadIdx.x;
  if (i >= (size_t)Nn * D) return;
  int n = (int)(i / D);
  float wsv = wselfp[wsidx];
  float v = agg[i] + dis[n] * dis[n] * wsv * hm[i];
  if (origin) v += origin[i];
  agg[i] = LEAKY(v);
}

// graclus pool with cluster = i//2: xp[c] = max(h[2c], h[2c+1]) elementwise
__global__ void pair_max_kernel(const float* __restrict__ h, float* __restrict__ xp,
                                int N2, int D) {
  size_t i = (size_t)blockIdx.x * blockDim.x + threadIdx.x;
  if (i >= (size_t)N2 * D) return;
  int n = (int)(i / D), d = (int)(i % D);
  float a = h[(size_t)(2 * n) * D + d];
  float b = h[(size_t)(2 * n + 1) * D + d];
  xp[i] = a > b ? a : b;
}

__global__ void global_max_kernel(const float* __restrict__ h1,
                                  const int* __restrict__ batch_p,
                                  float* __restrict__ g, int N2, int D) {
  size_t i = (size_t)blockIdx.x * blockDim.x + threadIdx.x;
  if (i >= (size_t)N2 * D) return;
  int n = (int)(i / D), d = (int)(i % D);
  atomMaxF(&g[(size_t)batch_p[n] * D + d], h1[i]);
}

__global__ void head_out_kernel(const float* __restrict__ ghid,
                                const float* __restrict__ W2, const float* __restrict__ b2,
                                float* __restrict__ out, int G, int Dh) {
  int r = blockIdx.x * blockDim.x + threadIdx.x;
  if (r >= G) return;
  float s = b2[0];
  for (int j = 0; j < Dh; ++j) s += ghid[(size_t)r * Dh + j] * W2[j];
  out[r] = s;
}

// ---------------------------------------------------------------------------
extern "C" void kernel_launch(void* const* d_in, const int* in_sizes, int n_in,
                              void* d_out, int out_size, void* d_ws, size_t ws_size,
                              hipStream_t stream) {
  const int D  = 128;
  const int N  = in_sizes[0] / D;
  const int E  = in_sizes[2] / 12;
  const int N2 = in_sizes[5];
  const int H  = in_sizes[15];   // l1_b1
  const int Dh = in_sizes[23];   // hd_b1
  const int G  = out_size;       // 64 graphs, 1 output each

  const float* x       = (const float*)d_in[0];
  const float* origin  = (const float*)d_in[1];
  const float* eattr   = (const float*)d_in[2];
  const int*   eidx    = (const int*)d_in[3];
  const int*   cluster = (const int*)d_in[4];
  const int*   batch_p = (const int*)d_in[5];
  const float *l0_W1 = (const float*)d_in[6],  *l0_b1 = (const float*)d_in[7];
  const float *l0_s  = (const float*)d_in[8],  *l0_t  = (const float*)d_in[9];
  const float *l0_W2 = (const float*)d_in[10], *l0_b2 = (const float*)d_in[11];
  const float *l0_We = (const float*)d_in[12], *l0_be = (const float*)d_in[13];
  const float *l1_W1 = (const float*)d_in[14], *l1_b1 = (const float*)d_in[15];
  const float *l1_s  = (const float*)d_in[16], *l1_t  = (const float*)d_in[17];
  const float *l1_W2 = (const float*)d_in[18], *l1_b2 = (const float*)d_in[19];
  const float *l1_We = (const float*)d_in[20], *l1_be = (const float*)d_in[21];
  const float *hd_W1 = (const float*)d_in[22], *hd_b1 = (const float*)d_in[23];
  const float *hd_s  = (const float*)d_in[24], *hd_t  = (const float*)d_in[25];
  const float *hd_W2 = (const float*)d_in[26], *hd_b2 = (const float*)d_in[27];

  const int* row = eidx;
  const int* col = eidx + E;

  // workspace layout (floats), with buffer reuse across phases
  const size_t nd   = (size_t)N * D;
  const size_t szA  = nd > (size_t)N2 * H ? nd : (size_t)N2 * H;
  float* p    = (float*)d_ws;
  float* bufA = p; p += szA;              // MLP hidden (layer0), then hid1 (layer1)
  float* bufB = p; p += nd;               // MLP out h (layer0), then h1 mlp out
  float* bufC = p; p += nd;               // aggregate -> post-combine features
  float* xp   = p; p += (size_t)N2 * D;   // pooled features
  float* dis0 = p; p += N;
  float* dis1 = p; p += N2;
  float* w0   = p; p += E;
  float* w1   = p; p += E;
  float* gbuf = p; p += (size_t)G * D;
  float* ghid = p; p += (size_t)G * Dh;
  float* wself= p; p += 2;

  auto cdiv = [](size_t a, size_t b) { return (int)((a + b - 1) / b); };
  auto gemm = [&](const float* A, const float* W, const float* b,
                  const float* sc, const float* tc, float* out,
                  int M, int K, int Nc, int flags) {
    dim3 block(128);                              // 4 waves -> 4 M-tiles/block
    dim3 grid(Nc / 64, cdiv(M, 64));              // each wave covers 64 cols
    gemm_wmma_kernel<<<grid, block, 0, stream>>>(A, W, b, sc, tc, out, M, K, Nc, flags);
  };

  // ---- degree / normalization / edge weights ----
  fill_kernel<<<cdiv(N, 256), 256, 0, stream>>>(dis0, 0.f, N);
  fill_kernel<<<cdiv(N2, 256), 256, 0, stream>>>(dis1, 0.f, N2);
  edge_deg_kernel<<<cdiv(E, 256), 256, 0, stream>>>(row, col, nullptr, dis0, E);
  edge_deg_kernel<<<cdiv(E, 256), 256, 0, stream>>>(row, col, cluster, dis1, E);
  dis_kernel<<<cdiv(N, 256), 256, 0, stream>>>(dis0, N);
  dis_kernel<<<cdiv(N2, 256), 256, 0, stream>>>(dis1, N2);
  edge_w_kernel<<<cdiv(E, 256), 256, 0, stream>>>(eattr, l0_We, l0_be, l1_We, l1_be, w0, w1, E);
  wself_kernel<<<1, 32, 0, stream>>>(l0_We, l0_be, l1_We, l1_be, wself);

  // ---- layer 0: MLP (two WMMA GEMMs) ----
  gemm(x,    l0_W1, l0_b1, l0_s, l0_t, bufA, N, D, D, /*bn+lrelu*/3);
  gemm(bufA, l0_W2, l0_b2, nullptr, nullptr, bufB, N, D, D, 0);

  // ---- layer 0: edge aggregation + self-loop + residual + leaky ----
  fill_kernel<<<cdiv(nd, 256), 256, 0, stream>>>(bufC, 0.f, nd);
  edge_agg_kernel<<<cdiv((size_t)E * 32, 256), 256, 0, stream>>>(
      bufB, row, col, w0, dis0, nullptr, bufC, E);
  combine_kernel<<<cdiv(nd, 256), 256, 0, stream>>>(
      bufC, bufB, origin, dis0, wself, 0, N, D);

  // ---- graclus max pool (cluster = i//2 pairs) ----
  pair_max_kernel<<<cdiv((size_t)N2 * D, 256), 256, 0, stream>>>(bufC, xp, N2, D);

  // ---- layer 1: MLP (128 -> 256 -> 128) ----
  gemm(xp,   l1_W1, l1_b1, l1_s, l1_t, bufA, N2, D, H, 3);
  gemm(bufA, l1_W2, l1_b2, nullptr, nullptr, bufB, N2, H, D, 0);

  // ---- layer 1: pooled-edge aggregation + self-loop + leaky ----
  fill_kernel<<<cdiv((size_t)N2 * D, 256), 256, 0, stream>>>(bufC, 0.f, (size_t)N2 * D);
  edge_agg_kernel<<<cdiv((size_t)E * 32, 256), 256, 0, stream>>>(
      bufB, row, col, w1, dis1, cluster, bufC, E);
  combine_kernel<<<cdiv((size_t)N2 * D, 256), 256, 0, stream>>>(
      bufC, bufB, nullptr, dis1, wself, 1, N2, D);

  // ---- global max pool over graphs ----
  fill_kernel<<<cdiv((size_t)G * D, 256), 256, 0, stream>>>(gbuf, -INFINITY, (size_t)G * D);
  global_max_kernel<<<cdiv((size_t)N2 * D, 256), 256, 0, stream>>>(bufC, batch_p, gbuf, N2, D);

  // ---- head MLP: [G,128] -> [G,64] (WMMA) -> [G,1] ----
  gemm(gbuf, hd_W1, hd_b1, hd_s, hd_t, ghid, G, D, Dh, 3);
  head_out_kernel<<<cdiv(G, 64), 64, 0, stream>>>(ghid, hd_W2, hd_b2, (float*)d_out, G, Dh);
}